// SpanPIIClassifier_42752104465048
// MI455X (gfx1250) — compile-verified
//
#include <hip/hip_runtime.h>

// CDNA5 / gfx1250 fused span-classifier:
//   logits[b,n] = relu(concat(lh[b,s0],lh[b,s1]) @ W1 + b1) @ W2 + b2
// Full-precision path via V_WMMA_F32_16X16X4_F32 (A,B = fp32).
// v3: 64-span M-block. Each K-step loads 4 A frags + 2 B frags and feeds
//     8 WMMAs -> 0.5 global dword + 0.5 ds b64 per WMMA.

typedef float v2f __attribute__((ext_vector_type(2)));
typedef float v8f __attribute__((ext_vector_type(8)));

#define B_     32
#define S_     512
#define H_     768
#define NSPAN  1024
#define KDIM   (2 * H_)        // 1536 reduction length
#define MT     64              // spans per block (four 16-row M tiles)
#define MTILES 4
#define KCH    192             // K chunk staged in LDS; 768 % 192 == 0
#define AST    (KCH + 4)       // LDS row stride (dwords): 196 % 64 == 4 -> conflict-free
#define NWAVE  8
#define NTILES (H_ / 16)       // 48 N tiles of W1
#define NPASS  3               // passes over K per block
#define TPP    2               // N tiles per wave per pass (8*3*2 = 48)

__global__ __launch_bounds__(256) void span_pii_kernel(
    const float* __restrict__ lh,   // (B, S, H)
    const float* __restrict__ W1,   // (2H, H) row-major
    const float* __restrict__ b1,   // (H)
    const float* __restrict__ W2,   // (H, 1)
    const float* __restrict__ b2,   // (1)
    const int*  __restrict__ spans, // (B, NSPAN, 2)
    float* __restrict__ out)        // (B, NSPAN)
{
    __shared__ __attribute__((aligned(16))) float A[MT * AST];   // 64 x 196 f32 = 50,176 B
    __shared__ int   rowbase[2][MT];
    __shared__ float partial[MT][NWAVE];

    const int tid  = threadIdx.x;
    const int lane = tid & 31;
    const int wave = tid >> 5;
    const int s0   = blockIdx.x * MT;     // 1024 % 64 == 0: block never crosses a batch

    if (tid < MT) {
        const int s  = s0 + tid;
        const int b  = s >> 10;                      // s / NSPAN
        const int st = spans[2 * s + 0];
        const int en = spans[2 * s + 1];
        rowbase[0][tid] = (b * S_ + st) * H_;        // h_start row (K 0..767)
        rowbase[1][tid] = (b * S_ + en) * H_;        // h_end   row (K 768..1535)
    }
    __syncthreads();

    const int laneN = lane & 15;          // N column within tile
    const int koff  = (lane >> 4) << 1;   // fragment K sub-offset: 0 (lanes 0-15), 2 (lanes 16-31)

    v8f lacc[MTILES];                     // running relu(h)*W2 partials, one per M tile
#pragma unroll
    for (int t = 0; t < MTILES; ++t)
#pragma unroll
        for (int i = 0; i < 8; ++i) lacc[t][i] = 0.0f;

    // Staging map: one quad of threads per A row, 12 float4 per thread per chunk.
    const int srow = tid >> 2;            // 0..63
    const int scol = tid & 3;

    for (int p = 0; p < NPASS; ++p) {
        // C accumulators: TPP N-tiles x MTILES M-tiles, initialized with b1 broadcast.
        v8f c[TPP][MTILES];
#pragma unroll
        for (int j = 0; j < TPP; ++j) {
            const float bv = b1[(wave * NPASS * TPP + p * TPP + j) * 16 + laneN];
#pragma unroll
            for (int t = 0; t < MTILES; ++t)
#pragma unroll
                for (int i = 0; i < 8; ++i) c[j][t][i] = bv;
        }

        for (int k0 = 0; k0 < KDIM; k0 += KCH) {
            // Stage 64 x KCH gathered f32 into LDS (float4, 64B-coalesced per quad).
            {
                const int base = (k0 < H_) ? rowbase[0][srow] + k0
                                           : rowbase[1][srow] + k0 - H_;
#pragma unroll
                for (int i = 0; i < KCH / 16; ++i) {        // 12 float4 per thread
                    const int c4 = (scol + (i << 2)) << 2;  // float offset in chunk
                    const float4 v = *reinterpret_cast<const float4*>(lh + base + c4);
                    *reinterpret_cast<float4*>(&A[srow * AST + c4]) = v;
                }
            }
            __syncthreads();

            const float* arow = &A[(lane & 15) * AST + koff];  // M rows 0-15; +16*AST per tile

            const int ncol0 = (wave * NPASS * TPP + p * TPP + 0) * 16 + laneN;
            const int ncol1 = (wave * NPASS * TPP + p * TPP + 1) * 16 + laneN;
            const float* bp0 = W1 + (size_t)(k0 + koff) * H_ + ncol0;
            const float* bp1 = W1 + (size_t)(k0 + koff) * H_ + ncol1;

#pragma unroll 2
            for (int kk = 0; kk < KCH; kk += 4) {
                const v2f a0 = *reinterpret_cast<const v2f*>(arow + kk);
                const v2f a1 = *reinterpret_cast<const v2f*>(arow + 16 * AST + kk);
                const v2f a2 = *reinterpret_cast<const v2f*>(arow + 32 * AST + kk);
                const v2f a3 = *reinterpret_cast<const v2f*>(arow + 48 * AST + kk);
                v2f bb0; bb0.x = bp0[0]; bb0.y = bp0[H_];
                v2f bb1; bb1.x = bp1[0]; bb1.y = bp1[H_];
                // 8 WMMAs per 4 A + 2 B fragment loads.
                c[0][0] = __builtin_amdgcn_wmma_f32_16x16x4_f32(false, a0, false, bb0, (short)0, c[0][0], false, false);
                c[0][1] = __builtin_amdgcn_wmma_f32_16x16x4_f32(false, a1, false, bb0, (short)0, c[0][1], false, false);
                c[0][2] = __builtin_amdgcn_wmma_f32_16x16x4_f32(false, a2, false, bb0, (short)0, c[0][2], false, false);
                c[0][3] = __builtin_amdgcn_wmma_f32_16x16x4_f32(false, a3, false, bb0, (short)0, c[0][3], false, false);
                c[1][0] = __builtin_amdgcn_wmma_f32_16x16x4_f32(false, a0, false, bb1, (short)0, c[1][0], false, false);
                c[1][1] = __builtin_amdgcn_wmma_f32_16x16x4_f32(false, a1, false, bb1, (short)0, c[1][1], false, false);
                c[1][2] = __builtin_amdgcn_wmma_f32_16x16x4_f32(false, a2, false, bb1, (short)0, c[1][2], false, false);
                c[1][3] = __builtin_amdgcn_wmma_f32_16x16x4_f32(false, a3, false, bb1, (short)0, c[1][3], false, false);
                bp0 += 4 * H_;
                bp1 += 4 * H_;
            }
            __syncthreads();  // protect LDS A before restage
        }

        // Fold this pass's N columns into the running logit partials.
#pragma unroll
        for (int j = 0; j < TPP; ++j) {
            const float w2v = W2[(wave * NPASS * TPP + p * TPP + j) * 16 + laneN];
#pragma unroll
            for (int t = 0; t < MTILES; ++t)
#pragma unroll
                for (int i = 0; i < 8; ++i)
                    lacc[t][i] += fmaxf(c[j][t][i], 0.0f) * w2v;
        }
    }

    // Reduce the 16 N-lanes inside each half-wave, then combine waves via LDS.
#pragma unroll
    for (int t = 0; t < MTILES; ++t) {
#pragma unroll
        for (int i = 0; i < 8; ++i) {
            float v = lacc[t][i];
            v += __shfl_xor(v, 1);
            v += __shfl_xor(v, 2);
            v += __shfl_xor(v, 4);
            v += __shfl_xor(v, 8);
            if (laneN == 0) {
                const int m = t * 16 + i + ((lane >> 4) << 3);
                partial[m][wave] = v;
            }
        }
    }
    __syncthreads();

    if (tid < MT) {
        float r = b2[0];
#pragma unroll
        for (int w = 0; w < NWAVE; ++w) r += partial[tid][w];
        out[s0 + tid] = r;
    }
}

extern "C" void kernel_launch(void* const* d_in, const int* in_sizes, int n_in,
                              void* d_out, int out_size, void* d_ws, size_t ws_size,
                              hipStream_t stream) {
    (void)in_sizes; (void)n_in; (void)d_ws; (void)ws_size; (void)out_size;
    const float* lh    = (const float*)d_in[0];  // last_hidden (32,512,768)
    const float* W1    = (const float*)d_in[1];  // (1536,768)
    const float* b1    = (const float*)d_in[2];  // (768)
    const float* W2    = (const float*)d_in[3];  // (768,1)
    const float* b2    = (const float*)d_in[4];  // (1)
    const int*   spans = (const int*)d_in[5];    // (32,1024,2) int32 per harness convention
    float*       out   = (float*)d_out;          // (32,1024)

    const int nblocks = (B_ * NSPAN) / MT;       // 512
    span_pii_kernel<<<nblocks, 256, 0, stream>>>(lh, W1, b1, W2, b2, spans, out);
}